// MultiHeadGATLayer_67559835566267
// MI455X (gfx1250) — compile-verified
//
#include <hip/hip_runtime.h>
#include <hip/hip_bf16.h>

// ---------------------------------------------------------------------------
// MultiHeadGAT on MI455X (gfx1250, wave32).
//
// Phase 1 (trivial): wh = h @ W^T + b  (2048x64), e_i/e_j per (node, head).
// Phase 2 (dominant): flash-style masked softmax + aggregation.
//   - 1 workgroup = one 16-row i-tile, 8 waves = 8 heads; adj tile staged in
//     LDS so the 16 MB adjacency matrix is streamed exactly once.
//   - P @ wh accumulated with V_WMMA_F32_16X16X4_F32 (fp32 WMMA, K=4/op).
//   - All masking is arithmetic (cndmask of constants * value): no exec
//     divergence in the hot loop, as WMMA requires EXEC all-ones anyway.
//   - LDS tiles stored j-contiguous per head/d with row stride 132 dwords:
//     one ds_load_b64 per operand pair, conflict-free half-wave bank walk.
// ---------------------------------------------------------------------------

typedef __attribute__((ext_vector_type(2))) float v2f;
typedef __attribute__((ext_vector_type(8))) float v8f;

#define NN      2048
#define IN_DIM  256
#define OUT_DIM 64
#define NHEADS  8
#define DH      8
#define JCHUNK  128   // j-tile per LDS stage
#define LSTR    132   // padded LDS row stride (dwords): 4-dword bank walk, b64-aligned

// ------------------------- Phase 1: wh, e_i, e_j ---------------------------
__global__ void gat_wh_kernel(const float* __restrict__ h,
                              const float* __restrict__ W_w,
                              const float* __restrict__ W_b,
                              const float* __restrict__ a_w,
                              float* __restrict__ wh,
                              float* __restrict__ e_i,
                              float* __restrict__ e_j) {
    __shared__ float hrow[IN_DIM];
    __shared__ float whs[OUT_DIM];
    const int n = blockIdx.x;
    const int t = threadIdx.x;              // 0..63

    ((float4*)hrow)[t] = ((const float4*)(h + (size_t)n * IN_DIM))[t];
    __syncthreads();

    const float4* wr4 = (const float4*)(W_w + (size_t)t * IN_DIM);
    const float4* hr4 = (const float4*)hrow;
    float acc = W_b[t];
#pragma unroll 8
    for (int k = 0; k < IN_DIM / 4; ++k) {
        float4 w = wr4[k];
        float4 x = hr4[k];
        acc = fmaf(w.x, x.x, acc);
        acc = fmaf(w.y, x.y, acc);
        acc = fmaf(w.z, x.z, acc);
        acc = fmaf(w.w, x.w, acc);
    }
    wh[(size_t)n * OUT_DIM + t] = acc;
    whs[t] = acc;
    __syncthreads();

    if (t < NHEADS) {                       // e_i[n][t] = wh_head[n,t,:] . aL
        float s = 0.f;
#pragma unroll
        for (int d = 0; d < DH; ++d) s = fmaf(whs[t * DH + d], a_w[d], s);
        e_i[(size_t)n * NHEADS + t] = s;
    } else if (t < 2 * NHEADS) {            // e_j[n][hh] = wh_head[n,hh,:] . aR
        const int hh = t - NHEADS;
        float s = 0.f;
#pragma unroll
        for (int d = 0; d < DH; ++d) s = fmaf(whs[hh * DH + d], a_w[DH + d], s);
        e_j[(size_t)n * NHEADS + hh] = s;
    }
}

// ---------------- Phase 2: masked softmax + WMMA aggregation ---------------
// Grid: 128 blocks (16 rows each) x 256 threads (8 waves = 8 heads).
__global__ __launch_bounds__(256)
void gat_att_kernel(const int*   __restrict__ adj,
                    const float* __restrict__ wh,
                    const float* __restrict__ e_i,
                    const float* __restrict__ e_j,
                    const float* __restrict__ a_b,
                    float* __restrict__ out) {
    __shared__ int   adjS[16 * LSTR];            // [i][j]   8.25 KB
    __shared__ float whT[OUT_DIM * LSTR];        // [d][j]  33.8  KB (transposed)
    __shared__ float ejT[NHEADS * LSTR];         // [h][j]   4.2  KB (transposed)
    __shared__ float pden[NHEADS][32];           //          1    KB

    const int tid  = threadIdx.x;
    const int hH   = tid >> 5;                   // wave id == head
    const int lane = tid & 31;
    const int i0   = blockIdx.x * 16;

    const int   iLoc  = lane & 15;               // A-matrix row (M)
    const int   nCol  = lane & 15;               // B/D column (N)
    const int   nColC = nCol & 7;                // aliased valid column
    const int   khalf = (lane >> 4) * 2;         // lanes>=16 hold K=2,3
    const float cmask = (nCol < DH) ? 1.0f : 0.0f;
    const float eib   = e_i[(size_t)(i0 + iLoc) * NHEADS + hH] + a_b[0];
    const int   dRow  = hH * DH + nColC;         // whT row for this lane

    v8f   acc  = {0.f, 0.f, 0.f, 0.f, 0.f, 0.f, 0.f, 0.f};
    float psum = 0.f;

    for (int chunk = 0; chunk < NN; chunk += JCHUNK) {
        __syncthreads();   // previous stage fully consumed before overwrite

        // adj tile: 16 rows x 128 cols, int4 loads, padded-row store
        for (int idx = tid; idx < 512; idx += 256) {
            const int r = idx >> 5;
            const int c = idx & 31;
            ((int4*)&adjS[r * LSTR])[c] =
                *(const int4*)(adj + (size_t)(i0 + r) * NN + chunk + c * 4);
        }
        // wh chunk transposed: whT[d][j] = wh[chunk+j][d]  (coalesced reads)
        for (int idx = tid; idx < JCHUNK * OUT_DIM; idx += 256) {
            const int j = idx >> 6;
            const int d = idx & 63;
            whT[d * LSTR + j] = wh[(size_t)(chunk + j) * OUT_DIM + d];
        }
        // e_j chunk transposed: ejT[h][j] = e_j[chunk+j][h]
        for (int idx = tid; idx < JCHUNK * NHEADS; idx += 256) {
            const int j  = idx >> 3;
            const int hh = idx & 7;
            ejT[hh * LSTR + j] = e_j[(size_t)(chunk + j) * NHEADS + hh];
        }

        if (chunk + JCHUNK < NN)   // warm L2 for the next adj slab
            __builtin_prefetch(adj + (size_t)(i0 + (tid >> 4)) * NN + chunk + JCHUNK, 0, 0);

        __syncthreads();

#pragma unroll
        for (int js = 0; js < JCHUNK; js += 4) {
            const int j0 = js + khalf;           // this lane-half's K pair
            const float2 ej = *(const float2*)&ejT[hH * LSTR + j0];   // b64
            const int2   ad = *(const int2*)  &adjS[iLoc * LSTR + j0];// b64
            const float2 wv = *(const float2*)&whT[dRow * LSTR + j0]; // b64

            float s0 = eib + ej.x;
            float s1 = eib + ej.y;
            s0 = (s0 >= 0.f) ? s0 : 0.2f * s0;   // leaky relu
            s1 = (s1 >= 0.f) ? s1 : 0.2f * s1;
            const float m0 = (ad.x > 0) ? 1.0f : 0.0f;  // cndmask, no branch
            const float m1 = (ad.y > 0) ? 1.0f : 0.0f;
            const float p0 = __expf(s0) * m0;
            const float p1 = __expf(s1) * m1;
            psum += p0 + p1;

            v2f A; A.x = p0;          A.y = p1;          // A[M=iLoc][K pair]
            v2f B; B.x = wv.x * cmask; B.y = wv.y * cmask; // B[K pair][N=nCol]
            acc = __builtin_amdgcn_wmma_f32_16x16x4_f32(
                false, A, false, B, (short)0, acc, false, false);
        }
    }

    // softmax denominator: row i = lanes i (K=0,1 half) + i+16 (K=2,3 half)
    pden[hH][lane] = psum;
    __syncthreads();

    const int rbase = (lane >> 4) * 8;           // D rows: v + 8*(lane>=16)
    if (nCol < DH) {
#pragma unroll
        for (int v = 0; v < 8; ++v) {
            const int   iL  = rbase + v;
            const float den = pden[hH][iL] + pden[hH][iL + 16];
            float o = acc[v] / den;
            o = (o > 0.f) ? o : (__expf(o) - 1.f);   // elu
            out[(size_t)hH * NN * DH + (size_t)(i0 + iL) * DH + nCol] = o;
        }
    }
}

// ---------------------------------------------------------------------------
extern "C" void kernel_launch(void* const* d_in, const int* in_sizes, int n_in,
                              void* d_out, int out_size, void* d_ws, size_t ws_size,
                              hipStream_t stream) {
    const float* h   = (const float*)d_in[0];
    const int*   adj = (const int*)  d_in[1];
    const float* W_w = (const float*)d_in[2];
    const float* W_b = (const float*)d_in[3];
    const float* a_w = (const float*)d_in[4];
    const float* a_b = (const float*)d_in[5];
    float* out = (float*)d_out;

    float* wh  = (float*)d_ws;                       // 2048*64  = 512 KB
    float* e_i = wh  + (size_t)NN * OUT_DIM;         // 2048*8   =  64 KB
    float* e_j = e_i + (size_t)NN * NHEADS;          // 2048*8   =  64 KB

    gat_wh_kernel<<<NN, 64, 0, stream>>>(h, W_w, W_b, a_w, wh, e_i, e_j);
    gat_att_kernel<<<NN / 16, 256, 0, stream>>>(adj, wh, e_i, e_j, a_b, out);
}